// LocalVoxelEncoder_4587025072486
// MI455X (gfx1250) — compile-verified
//
#include <hip/hip_runtime.h>

// ---------------------------------------------------------------------------
// LocalVoxelEncoder fused kernel for gfx1250 (CDNA5, wave32).
//   conv3d(1->128, 3x3x3, SAME) + bias + relu  ==> im2col GEMM [M x 27]x[27 x 128]
//   scatter_mean onto 128x128 plane            ==> mean over h (count==64), unique
//                                                  cell per (d,w), rest zero.
// GEMM runs on V_WMMA_F32_16X16X4_F32 (f32 in, f32 acc), K padded 27->28 = 7 WMMAs.
// ---------------------------------------------------------------------------

typedef __attribute__((ext_vector_type(2))) float v2f;
typedef __attribute__((ext_vector_type(8))) float v8f;

#define CDIM   128
#define RESO   128
#define DSZ    64
#define HSZ    64
#define WSZ    64
#define HCHUNK 16                 // h rows per workgroup
#define KPAD   28                 // 27 taps padded to 28 (7 chunks of K=4)

// slab in LDS: [3 d][HCHUNK+2 h][WSZ+2 w], halo pre-zeroed
#define XS_H   (HCHUNK + 2)
#define XS_W   (WSZ + 2)
#define XS_N   (3 * XS_H * XS_W)

// plane cell mapping, identical math to the reference:
//   c = linspace(-0.5, 0.5, 64)[i];  t = c/1.101 + 0.5; clamp; floor(t*128)
__device__ __forceinline__ int plane_cell(int i) {
    float c = -0.5f + (float)i * (1.0f / 63.0f);
    float t = c / 1.101f + 0.5f;
    if (t >= 1.0f) t = 1.0f - 1e-5f;
    if (t <  0.0f) t = 0.0f;
    return (int)(t * 128.0f);
}

// im2col element A[w_row][k] for the current (hl) row, from the LDS slab.
// k = dd*9 + dh*3 + dw  (OIDHW weight order, cross-correlation, pad=1)
__device__ __forceinline__ float aval(const float* xs, int k, int hl, int wrow) {
    if (k >= 27) return 0.0f;            // K padding
    int dd = k / 9;
    int r  = k - dd * 9;
    int dh = r / 3;
    int dw = r - dh * 3;
    return xs[(dd * XS_H + (hl + dh)) * XS_W + (wrow + dw)];
}

__global__ void lve_zero_kernel(float* __restrict__ out, int n) {
    int i = blockIdx.x * blockDim.x + threadIdx.x;
    if (i < n) out[i] = 0.0f;
}

__global__ __launch_bounds__(256)
void lve_conv_scatter_kernel(const float* __restrict__ x,
                             const float* __restrict__ wgt,   // [128][27]
                             const float* __restrict__ bias,  // [128]
                             float* __restrict__ out) {       // [2][128][128][128]
    __shared__ float xs[XS_N];           // x slab with halo      (~14.3 KB)
    __shared__ float Bs[KPAD * CDIM];    // weights, k-major      (~14.3 KB)

    const int b   = blockIdx.z;          // batch   0..1
    const int d   = blockIdx.y;          // depth   0..63
    const int h0  = blockIdx.x * HCHUNK; // h chunk base
    const int tid = threadIdx.x;

    // ---- stage weights into LDS as Bs[k][n], zero-pad k=27 ----
    for (int e = tid; e < KPAD * CDIM; e += 256) {
        int k = e >> 7;                  // /128
        int n = e & (CDIM - 1);
        Bs[e] = (k < 27) ? wgt[n * 27 + k] : 0.0f;
    }
    // ---- stage x slab with halo (zero padded at volume borders) ----
    for (int e = tid; e < XS_N; e += 256) {
        int dd = e / (XS_H * XS_W);
        int r  = e - dd * (XS_H * XS_W);
        int hh = r / XS_W;
        int ww = r - hh * XS_W;
        int gd = d  + dd - 1;
        int gh = h0 + hh - 1;
        int gw = ww - 1;
        float v = 0.0f;
        if ((unsigned)gd < (unsigned)DSZ &&
            (unsigned)gh < (unsigned)HSZ &&
            (unsigned)gw < (unsigned)WSZ)
            v = x[((b * DSZ + gd) * HSZ + gh) * WSZ + gw];
        xs[e] = v;
    }
    __syncthreads();

    const int wave = tid >> 5;           // 8 waves -> 8 channel tiles of 16
    const int lane = tid & 31;
    const int hi   = lane >> 4;          // lane half: 0 | 1
    const int nloc = lane & 15;
    const int ch   = wave * 16 + nloc;   // this lane's channel (N index)

    // ---- B fragments (persist across whole kernel) ----
    // f32 4x16 B operand layout (mirror of the documented 16x4 A layout):
    //   VGPR0: lanes 0-15 = B[k0+0][n], lanes 16-31 = B[k0+2][n]
    //   VGPR1: lanes 0-15 = B[k0+1][n], lanes 16-31 = B[k0+3][n]
    v2f bf[7];
    #pragma unroll
    for (int kc = 0; kc < 7; ++kc) {
        int k0 = kc * 4 + hi * 2;
        bf[kc].x = Bs[(k0 + 0) * CDIM + ch];
        bf[kc].y = Bs[(k0 + 1) * CDIM + ch];
    }
    const float bv    = bias[ch];
    const int   celld = plane_cell(d);

    // ---- 4 w-tiles of 16 voxels; reduce over the 16 h rows of this chunk ----
    for (int wt = 0; wt < 4; ++wt) {
        const int wrow = wt * 16 + nloc; // A row (M index) held by this lane
        v8f acc = {};                    // post-ReLU h-partial sums
        for (int hl = 0; hl < HCHUNK; ++hl) {
            v8f c = {};
            #pragma unroll
            for (int kc = 0; kc < 7; ++kc) {
                int k0 = kc * 4 + hi * 2;
                v2f a;
                a.x = aval(xs, k0 + 0, hl, wrow);
                a.y = aval(xs, k0 + 1, hl, wrow);
                // D = A(16x4) * B(4x16) + C   (K accumulation chained over kc)
                c = __builtin_amdgcn_wmma_f32_16x16x4_f32(
                        /*neg_a=*/false, a, /*neg_b=*/false, bf[kc],
                        /*c_mod=*/(short)0, c, /*reuse_a=*/false, /*reuse_b=*/false);
            }
            // bias + relu BEFORE the h reduction (matches reference order)
            #pragma unroll
            for (int v = 0; v < 8; ++v) {
                float t = c[v] + bv;
                acc[v] += (t > 0.0f) ? t : 0.0f;
            }
        }
        // C/D layout: vgpr v, lane -> (M = v + 8*hi, N = nloc)
        // out[b][ch][cell(w)][cell(d)] += acc/64  (count per populated cell is 64)
        #pragma unroll
        for (int v = 0; v < 8; ++v) {
            int wr    = wt * 16 + v + 8 * hi;
            int cellw = plane_cell(wr);
            float* p  = out + (((size_t)b * CDIM + ch) * RESO + cellw) * RESO + celld;
            unsafeAtomicAdd(p, acc[v] * (1.0f / 64.0f));   // global_atomic_add_f32
        }
    }
}

extern "C" void kernel_launch(void* const* d_in, const int* in_sizes, int n_in,
                              void* d_out, int out_size, void* d_ws, size_t ws_size,
                              hipStream_t stream) {
    const float* x  = (const float*)d_in[0];   // (2,64,64,64)
    const float* cw = (const float*)d_in[1];   // (128,1,3,3,3) -> [128][27]
    const float* cb = (const float*)d_in[2];   // (128,)
    float* out = (float*)d_out;                // (2,128,128,128)

    // 1) zero the output plane (harness poisons d_out; empty cells must be 0)
    lve_zero_kernel<<<(out_size + 255) / 256, 256, 0, stream>>>(out, out_size);

    // 2) fused conv + relu + h-mean scatter
    dim3 grid(HSZ / HCHUNK, DSZ, 2);           // (4, 64, 2) = 512 workgroups
    lve_conv_scatter_kernel<<<grid, 256, 0, stream>>>(x, cw, cb, out);
}